// TimeDomainCochleagram_78305843740706
// MI455X (gfx1250) — compile-verified
//
#include <hip/hip_runtime.h>
#include <hip/hip_bf16.h>

// ---------------- problem constants ----------------
constexpr int BATCH_N = 8;
constexpr int CH      = 40;
constexpr int TN      = 96000;
constexpr int DSF     = 24;
constexpr int TOUT    = TN / DSF;        // 4000
constexpr int CL      = 768;             // chunk length, = 32 * 24
constexpr int NCH     = TN / CL;         // 125 chunks
constexpr int OUT_PER_CHUNK = CL / DSF;  // 32

typedef __attribute__((ext_vector_type(2))) float v2f;
typedef __attribute__((ext_vector_type(8))) float v8f;

// ---------------- per-channel coefficients ----------------
struct Coefs { float n00, n01, a0, a12, a13, a14, d1, d2; };

__device__ __forceinline__ Coefs load_coefs(const float* __restrict__ fc, int c) {
    const float* f = fc + c * 10;
    Coefs k;
    float a0 = f[0], g = f[9];
    k.n00 = a0 / g; k.n01 = f[1] / g;
    k.a0 = a0; k.a12 = f[2]; k.a13 = f[3]; k.a14 = f[4];
    k.d1 = f[7]; k.d2 = f[8];
    return k;
}

// One time-step of the 4-stage cascade (transposed DF-II, a2 == 0).
// State vector order q: [s1_0..s1_3, s2_0..s2_3].
__device__ __forceinline__ float fstep(const Coefs& K, float x,
                                       float s1[4], float s2[4]) {
    float y0 = fmaf(K.n00, x, s1[0]);
    float t0 = fmaf(K.n01, x, fmaf(-K.d1, y0, s2[0]));
    float u0 = -K.d2 * y0;
    float y1 = fmaf(K.a0, y0, s1[1]);
    float t1 = fmaf(K.a12, y0, fmaf(-K.d1, y1, s2[1]));
    float u1 = -K.d2 * y1;
    float y2 = fmaf(K.a0, y1, s1[2]);
    float t2 = fmaf(K.a13, y1, fmaf(-K.d1, y2, s2[2]));
    float u2 = -K.d2 * y2;
    float y3 = fmaf(K.a0, y2, s1[3]);
    float t3 = fmaf(K.a14, y2, fmaf(-K.d1, y3, s2[3]));
    float u3 = -K.d2 * y3;
    s1[0]=t0; s1[1]=t1; s1[2]=t2; s1[3]=t3;
    s2[0]=u0; s2[1]=u1; s2[2]=u2; s2[3]=u3;
    return y3;
}

// ---------------- Kernel A: zero-state chunk end-states p ----------------
// thread -> (b, k, c) with c fastest: threads of one (b, k) stream the SAME x
// chunk (broadcast loads).
__global__ __launch_bounds__(256)
void k_chunk_endstate(const float* __restrict__ x, const float* __restrict__ fc,
                      float* __restrict__ p_ws) {
    int tid = blockIdx.x * blockDim.x + threadIdx.x;
    if (tid >= BATCH_N * CH * NCH) return;
    int c = tid % CH;
    int k = (tid / CH) % NCH;
    int b = tid / (CH * NCH);
    Coefs K = load_coefs(fc, c);
    float s1[4] = {0,0,0,0}, s2[4] = {0,0,0,0};
    const float4* xp = (const float4*)(x + b * TN + k * CL);
    #pragma unroll 4
    for (int i = 0; i < CL / 4; ++i) {
        float4 xv = xp[i];
        fstep(K, xv.x, s1, s2); fstep(K, xv.y, s1, s2);
        fstep(K, xv.z, s1, s2); fstep(K, xv.w, s1, s2);
    }
    float4* p4 = (float4*)(p_ws + ((b * CH + c) * NCH + k) * 8);
    p4[0] = make_float4(s1[0], s1[1], s1[2], s1[3]);
    p4[1] = make_float4(s2[0], s2[1], s2[2], s2[3]);
}

// ---------------- Kernel H: A^CL per channel (8x8, row-major) -------------
__global__ __launch_bounds__(64)
void k_matpow(const float* __restrict__ fc, float* __restrict__ AL) {
    int tid = blockIdx.x * blockDim.x + threadIdx.x;
    if (tid >= CH * 8) return;
    int c = tid % CH;
    int j = tid / CH;               // unit-state column
    Coefs K = load_coefs(fc, c);
    float s1[4] = {0,0,0,0}, s2[4] = {0,0,0,0};
    if (j < 4) s1[j] = 1.0f; else s2[j - 4] = 1.0f;
    for (int t = 0; t < CL; ++t) fstep(K, 0.0f, s1, s2);
    #pragma unroll
    for (int i = 0; i < 4; ++i) AL[c * 64 + i * 8 + j]       = s1[i];
    #pragma unroll
    for (int i = 0; i < 4; ++i) AL[c * 64 + (i + 4) * 8 + j] = s2[i];
}

// ---------------- Kernel B: inter-chunk scan via WMMA --------------------
// One wave per channel pair. 16x16 block-diagonal A^CL (2 channels), N =
// batches (8 used + 8 mirrored), K = 16 stacked states as 4 chained
// v_wmma_f32_16x16x4_f32. B operand rebuilt from the D accumulator with
// half-wave shuffles (ds_bpermute).
__global__ __launch_bounds__(32)
void k_interchunk_scan(const float* __restrict__ AL,
                       const float* __restrict__ p_ws,
                       float* __restrict__ s0_ws) {
    const int lane = threadIdx.x;
    const int n  = lane & 15;           // N column (batch)
    const int hl = lane >> 4;           // lane half
    const int c0 = blockIdx.x * 2;
    const int ch = c0 + hl;             // channel owned by this half in C/D
    const int nb = n & 7;               // clamped batch for mirrored columns

    // A operand: rows m = lane&15 span both channels (block-diagonal);
    // lane half selects K offset. Slice kb holds K columns 4kb..4kb+3.
    v2f a[4];
    #pragma unroll
    for (int kb = 0; kb < 4; ++kb) {
        float av[2];
        #pragma unroll
        for (int v = 0; v < 2; ++v) {
            int col = kb * 4 + hl * 2 + v;
            float val = 0.0f;
            if (n < 8 && col < 8)        val = AL[c0 * 64 + n * 8 + col];
            else if (n >= 8 && col >= 8) val = AL[(c0 + 1) * 64 + (n - 8) * 8 + (col - 8)];
            av[v] = val;
        }
        a[kb].x = av[0]; a[kb].y = av[1];
    }

    // Running state in C/D layout: st[r] = state r of channel ch, batch n.
    float st[8];
    #pragma unroll
    for (int r = 0; r < 8; ++r) st[r] = 0.0f;

    for (int k = 0; k < NCH; ++k) {
        // publish pre-chunk state s0[b][ch][k] (real batches only)
        if (n < 8) {
            float4* dst = (float4*)(s0_ws + ((n * CH + ch) * NCH + k) * 8);
            dst[0] = make_float4(st[0], st[1], st[2], st[3]);
            dst[1] = make_float4(st[4], st[5], st[6], st[7]);
        }
        // accumulator seeded with p[k]
        const float4* pp = (const float4*)(p_ws + ((nb * CH + ch) * NCH + k) * 8);
        float4 p0 = pp[0], p1 = pp[1];
        v8f acc;
        acc[0]=p0.x; acc[1]=p0.y; acc[2]=p0.z; acc[3]=p0.w;
        acc[4]=p1.x; acc[5]=p1.y; acc[6]=p1.z; acc[7]=p1.w;

        // 4 chained K-slices: acc += Abd[:,4kb:4kb+4] * S[4kb:4kb+4,:]
        #pragma unroll
        for (int kb = 0; kb < 4; ++kb) {
            const int srcLane = n + 16 * (kb >> 1);
            float blo0 = __shfl(st[(kb * 4 + 0) & 7], srcLane, 32);
            float blo1 = __shfl(st[(kb * 4 + 1) & 7], srcLane, 32);
            float bhi0 = __shfl(st[(kb * 4 + 2) & 7], srcLane, 32);
            float bhi1 = __shfl(st[(kb * 4 + 3) & 7], srcLane, 32);
            v2f bb;
            bb.x = hl ? bhi0 : blo0;
            bb.y = hl ? bhi1 : blo1;
            acc = __builtin_amdgcn_wmma_f32_16x16x4_f32(
                false, a[kb], false, bb, (short)0, acc, false, false);
        }
        #pragma unroll
        for (int r = 0; r < 8; ++r) st[r] = acc[r];
    }
}

// ---------------- Kernel C: exact chunk replay + ReLU + pool -------------
__global__ __launch_bounds__(256)
void k_final(const float* __restrict__ x, const float* __restrict__ fc,
             const float* __restrict__ s0_ws, float* __restrict__ out) {
    int tid = blockIdx.x * blockDim.x + threadIdx.x;
    if (tid >= BATCH_N * CH * NCH) return;
    int c = tid % CH;
    int k = (tid / CH) % NCH;
    int b = tid / (CH * NCH);
    Coefs K = load_coefs(fc, c);
    const float4* sp = (const float4*)(s0_ws + ((b * CH + c) * NCH + k) * 8);
    float4 sa = sp[0], sb = sp[1];
    float s1[4] = {sa.x, sa.y, sa.z, sa.w};
    float s2[4] = {sb.x, sb.y, sb.z, sb.w};
    const float4* xp = (const float4*)(x + b * TN + k * CL);
    float* op = out + (b * CH + c) * TOUT + k * OUT_PER_CHUNK;
    int i4 = 0;
    for (int o = 0; o < OUT_PER_CHUNK; ++o) {
        float acc = 0.0f;
        #pragma unroll
        for (int g = 0; g < DSF / 4; ++g) {
            float4 xv = xp[i4++];
            acc += fmaxf(fstep(K, xv.x, s1, s2), 0.0f);
            acc += fmaxf(fstep(K, xv.y, s1, s2), 0.0f);
            acc += fmaxf(fstep(K, xv.z, s1, s2), 0.0f);
            acc += fmaxf(fstep(K, xv.w, s1, s2), 0.0f);
        }
        op[o] = acc * (1.0f / DSF);
    }
}

// ---------------- launcher ----------------
extern "C" void kernel_launch(void* const* d_in, const int* in_sizes, int n_in,
                              void* d_out, int out_size, void* d_ws, size_t ws_size,
                              hipStream_t stream) {
    const float* x  = (const float*)d_in[0];   // (8, 96000) f32
    const float* fc = (const float*)d_in[1];   // (40, 10)   f32
    float* out = (float*)d_out;                // (8, 40, 4000) f32
    float* ws  = (float*)d_ws;

    float* p_ws  = ws;                                   // 8*40*125*8 = 320000 f
    float* s0_ws = ws + (size_t)BATCH_N * CH * NCH * 8;  // 320000 f
    float* AL    = s0_ws + (size_t)BATCH_N * CH * NCH * 8; // 40*64 f

    const int total = BATCH_N * CH * NCH;   // 40000 chunk-threads

    k_chunk_endstate<<<(total + 255) / 256, 256, 0, stream>>>(x, fc, p_ws);
    k_matpow<<<(CH * 8 + 63) / 64, 64, 0, stream>>>(fc, AL);
    k_interchunk_scan<<<CH / 2, 32, 0, stream>>>(AL, p_ws, s0_ws);
    k_final<<<(total + 255) / 256, 256, 0, stream>>>(x, fc, s0_ws, out);
}